// DC_21921513078817
// MI455X (gfx1250) — compile-verified
//
#include <hip/hip_runtime.h>
#include <hip/hip_bf16.h>

#define N_NODES 50000
#define E_EDGES 800000
#define IN_C    128
#define HID2    256
#define HID     128
#define M_TILES (N_NODES / 16)   // 3125, exact

typedef float v2f __attribute__((ext_vector_type(2)));
typedef float v8f __attribute__((ext_vector_type(8)));

// ---------------------------------------------------------------------------
// WMMA f32 GEMM: C[M,N] = A[M,K] @ B[K,N]; row-major, M=50000 (16-aligned),
// N in {128,256} (64-aligned), K in {128,256} (4-aligned).
// Each wave computes a 16x64 strip via 4 accumulators of V_WMMA_F32_16X16X4_F32.
// ---------------------------------------------------------------------------
__global__ void gcn_gemm_wmma_f32(const float* __restrict__ A,
                                  const float* __restrict__ B,
                                  float* __restrict__ C,
                                  int m_tiles, int N, int K) {
    const int lane = threadIdx.x & 31;
    const int wave = threadIdx.x >> 5;
    const int col_tiles = N >> 6;                    // 64-col groups
    const int t = blockIdx.x * 8 + wave;             // wave-uniform
    if (t >= m_tiles * col_tiles) return;            // uniform per wave -> EXEC all-1s
    const int rt = t / col_tiles;
    const int ct = t % col_tiles;
    const int row0 = rt * 16;
    const int col0 = ct * 64;

    const int m  = lane & 15;          // A row within tile / B column within tile
    const int kh = (lane >> 4) << 1;   // 0 for lanes 0-15, 2 for lanes 16-31

    v8f acc0 = {}, acc1 = {}, acc2 = {}, acc3 = {};
    const float* arow = A + (size_t)(row0 + m) * K + kh;

    for (int k0 = 0; k0 < K; k0 += 4) {
        // A tile 16x4: lane holds {A[m][k0+kh], A[m][k0+kh+1]}
        v2f a = *(const v2f*)(arow + k0);
        // B tiles 4x16: lane holds column (col0 + 16*j + m), K pair (k0+kh, k0+kh+1)
        const float* brow = B + (size_t)(k0 + kh) * N + col0 + m;
        v2f b0 = { brow[0],  brow[N] };
        v2f b1 = { brow[16], brow[N + 16] };
        v2f b2 = { brow[32], brow[N + 32] };
        v2f b3 = { brow[48], brow[N + 48] };
        acc0 = __builtin_amdgcn_wmma_f32_16x16x4_f32(false, a, false, b0, (short)0, acc0, false, false);
        acc1 = __builtin_amdgcn_wmma_f32_16x16x4_f32(false, a, false, b1, (short)0, acc1, false, false);
        acc2 = __builtin_amdgcn_wmma_f32_16x16x4_f32(false, a, false, b2, (short)0, acc2, false, false);
        acc3 = __builtin_amdgcn_wmma_f32_16x16x4_f32(false, a, false, b3, (short)0, acc3, false, false);
    }

    // D layout: VGPR j -> row = row0 + (lane>=16 ? 8 : 0) + j, col = col0 + (lane&15)
    const int srow = row0 + ((lane >> 4) << 3);
    const int scol = col0 + (lane & 15);
    #pragma unroll
    for (int j = 0; j < 8; ++j) {
        float* crow = C + (size_t)(srow + j) * N + scol;
        crow[0]  = acc0[j];
        crow[16] = acc1[j];
        crow[32] = acc2[j];
        crow[48] = acc3[j];
    }
}

// ---------------------------------------------------------------------------
// Elementwise / graph kernels
// ---------------------------------------------------------------------------
__global__ void gcn_zero_f32(float* __restrict__ p, int n) {
    int i = blockIdx.x * blockDim.x + threadIdx.x;
    if (i < n) p[i] = 0.0f;
}

__global__ void gcn_degree(const int* __restrict__ dst, float* __restrict__ deg, int E) {
    int e = blockIdx.x * blockDim.x + threadIdx.x;
    if (e < E) atomicAdd(&deg[dst[e]], 1.0f);
}

// in place: deg -> dinv = rsqrt(deg + 1)
__global__ void gcn_dinv(float* __restrict__ deg, int n) {
    int i = blockIdx.x * blockDim.x + threadIdx.x;
    if (i < n) deg[i] = rsqrtf(deg[i] + 1.0f);
}

// acc[i,c] = h[i,c] * dinv[i]^2   (self-loop term, also initializes acc)
__global__ void gcn_self_init(const float* __restrict__ h, const float* __restrict__ dinv,
                              float* __restrict__ acc, int total, int C) {
    int i = blockIdx.x * blockDim.x + threadIdx.x;
    if (i < total) {
        float d = dinv[i / C];
        acc[i] = h[i] * d * d;
    }
}

// one block per edge, one thread per channel: acc[dst] += h[src] * dinv[src]*dinv[dst]
__global__ void gcn_scatter(const float* __restrict__ h,
                            const int* __restrict__ src, const int* __restrict__ dst,
                            const float* __restrict__ dinv,
                            float* __restrict__ acc, int E, int C) {
    const int c = threadIdx.x;
    const int e = blockIdx.x;
    if (e >= E) return;
    const int s = src[e];
    const int d = dst[e];
    const float norm = dinv[s] * dinv[d];
    atomicAdd(&acc[(size_t)d * C + c], h[(size_t)s * C + c] * norm);
}

__global__ void gcn_bias_relu(const float* __restrict__ acc, const float* __restrict__ bias,
                              float* __restrict__ out, int total, int C) {
    int i = blockIdx.x * blockDim.x + threadIdx.x;
    if (i < total) {
        float v = acc[i] + bias[i % C];
        out[i] = v > 0.0f ? v : 0.0f;
    }
}

// ---------------------------------------------------------------------------
// Launch
// ---------------------------------------------------------------------------
extern "C" void kernel_launch(void* const* d_in, const int* in_sizes, int n_in,
                              void* d_out, int out_size, void* d_ws, size_t ws_size,
                              hipStream_t stream) {
    const float* x_a    = (const float*)d_in[0];
    const int*   edge_a = (const int*)d_in[1];
    const float* x_b    = (const float*)d_in[2];
    const int*   edge_b = (const int*)d_in[3];
    const float* W1     = (const float*)d_in[4];
    const float* b1     = (const float*)d_in[5];
    const float* W2     = (const float*)d_in[6];
    const float* b2     = (const float*)d_in[7];
    float* out = (float*)d_out;

    float* ws      = (float*)d_ws;
    float* buf_h   = ws;                                 // 50000*256 floats
    float* buf_acc = ws + (size_t)N_NODES * HID2;        // 50000*256 floats
    float* dinv    = ws + (size_t)2 * N_NODES * HID2;    // 50000 floats

    const int tot1 = N_NODES * HID2;                     // 12.8M
    const int tot2 = N_NODES * HID;                      // 6.4M
    const int gemm1_blocks = (M_TILES * (HID2 / 64) + 7) / 8;   // 12500 waves
    const int gemm2_blocks = (M_TILES * (HID  / 64) + 7) / 8;   // 6250 waves

    for (int g = 0; g < 2; ++g) {
        const float* x    = g ? x_b    : x_a;
        const int*   edge = g ? edge_b : edge_a;
        const int*   src  = edge;             // row 0 of [2,E]
        const int*   dst  = edge + E_EDGES;   // row 1 of [2,E]
        float* zout = out + (size_t)g * N_NODES * HID;

        // normalization: dinv = rsqrt(deg_in + 1)
        gcn_zero_f32<<<(N_NODES + 255) / 256, 256, 0, stream>>>(dinv, N_NODES);
        gcn_degree<<<(E_EDGES + 255) / 256, 256, 0, stream>>>(dst, dinv, E_EDGES);
        gcn_dinv<<<(N_NODES + 255) / 256, 256, 0, stream>>>(dinv, N_NODES);

        // ---- layer 1: h1 = x @ W1 ; agg ; relu(agg + b1) -> buf_acc ----
        gcn_gemm_wmma_f32<<<gemm1_blocks, 256, 0, stream>>>(x, W1, buf_h, M_TILES, HID2, IN_C);
        gcn_self_init<<<(tot1 + 255) / 256, 256, 0, stream>>>(buf_h, dinv, buf_acc, tot1, HID2);
        gcn_scatter<<<E_EDGES, HID2, 0, stream>>>(buf_h, src, dst, dinv, buf_acc, E_EDGES, HID2);
        gcn_bias_relu<<<(tot1 + 255) / 256, 256, 0, stream>>>(buf_acc, b1, buf_acc, tot1, HID2);

        // ---- layer 2: h2 = x1 @ W2 ; agg ; relu(agg + b2) -> d_out ----
        float* h2   = buf_h;                              // 50000*128 (first half)
        float* acc2 = buf_h + (size_t)N_NODES * HID;      // 50000*128 (second half)
        gcn_gemm_wmma_f32<<<gemm2_blocks, 256, 0, stream>>>(buf_acc, W2, h2, M_TILES, HID, HID2);
        gcn_self_init<<<(tot2 + 255) / 256, 256, 0, stream>>>(h2, dinv, acc2, tot2, HID);
        gcn_scatter<<<E_EDGES, HID, 0, stream>>>(h2, src, dst, dinv, acc2, E_EDGES, HID);
        gcn_bias_relu<<<(tot2 + 255) / 256, 256, 0, stream>>>(acc2, b2, zout, tot2, HID);
    }
}